// GCN_44624710205614
// MI455X (gfx1250) — compile-verified
//
#include <hip/hip_runtime.h>
#include <math.h>

typedef __attribute__((ext_vector_type(2))) float v2f;
typedef __attribute__((ext_vector_type(8))) float v8f;

#define NFEAT  128
#define LDSPAD 132   // 128 + 4 pad: column reads hit distinct LDS banks

// ---------------------------------------------------------------------------
// Dense GEMM with bias: H[m, n] = sum_k X[m,k] * W[k,n] + B[n]
// One block = 16 output rows x 128 cols. 8 waves, each wave owns a 16x16
// tile and iterates K=128 in 32 steps of V_WMMA_F32_16X16X4_F32.
// ---------------------------------------------------------------------------
__global__ __launch_bounds__(256) void gcn_gemm_bias(
    const float* __restrict__ X, const float* __restrict__ W,
    const float* __restrict__ Bv, float* __restrict__ H, int nnodes)
{
    __shared__ float xs[16 * LDSPAD];

    const int tid = threadIdx.x;
    const int m0  = blockIdx.x * 16;

    // Stage the 16x128 fp32 A-tile into LDS (8 floats per thread).
    #pragma unroll
    for (int i = 0; i < 8; ++i) {
        int idx = tid + i * 256;
        int r = idx >> 7;       // row 0..15
        int c = idx & 127;      // col 0..127
        int gr = m0 + r;
        xs[r * LDSPAD + c] = (gr < nnodes) ? X[(size_t)gr * NFEAT + c] : 0.0f;
    }
    __syncthreads();

    const int wave = tid >> 5;
    const int lane = tid & 31;
    const int nn   = lane & 15;        // N (B/C/D) or M (A) within tile
    const int kb   = (lane >> 4) << 1; // K sub-offset for this half-wave: 0 or 2
    const int n0   = wave * 16;        // this wave's output-column base

    // C init: bias depends only on N -> same value in all 8 acc VGPRs.
    v8f acc;
    {
        float bias = Bv[n0 + nn];
        #pragma unroll
        for (int r = 0; r < 8; ++r) acc[r] = bias;
    }

    // A(16x4 f32) layout: lanes 0-15 hold K = k+0 / k+1, lanes 16-31 K = k+2 / k+3.
    const float* xrow = &xs[nn * LDSPAD + kb];
    // B(4x16 f32) mirrors A with n in place of m; W is row-major [128x128].
    const float* wcol = &W[(size_t)kb * NFEAT + n0 + nn];

    #pragma unroll
    for (int k = 0; k < NFEAT; k += 4) {
        v2f a, b;
        a.x = xrow[k];                        // X[m, k+kb]     (ds_load_2addr_b64)
        a.y = xrow[k + 1];                    // X[m, k+kb+1]
        b.x = wcol[(size_t)k * NFEAT];        // W[k+kb,   n0+nn]
        b.y = wcol[(size_t)(k + 1) * NFEAT];  // W[k+kb+1, n0+nn]
        acc = __builtin_amdgcn_wmma_f32_16x16x4_f32(
            /*neg_a=*/false, a, /*neg_b=*/false, b,
            /*c_mod=*/(short)0, acc, /*reuse_a=*/false, /*reuse_b=*/false);
    }

    // D layout: VGPR r -> M = r (lanes 0-15) or r+8 (lanes 16-31); N = lane&15.
    const int mh = (lane >> 4) * 8;
    #pragma unroll
    for (int r = 0; r < 8; ++r) {
        int gm = m0 + r + mh;
        if (gm < nnodes) H[(size_t)gm * NFEAT + n0 + nn] = acc[r];
    }
}

// ---------------------------------------------------------------------------
// Edge scatter: AGG[rows[e], :] += vals[e] * H[cols[e], :]
// One wave per edge; each lane moves a float4 and issues 4 hardware
// GLOBAL_ATOMIC_ADD_F32 ops (no-return -> STOREcnt, fire-and-forget).
// ---------------------------------------------------------------------------
__global__ __launch_bounds__(256) void gcn_scatter(
    const float* __restrict__ H, const int* __restrict__ rows,
    const int* __restrict__ cols, const float* __restrict__ vals,
    float* __restrict__ AGG, int nedges)
{
    int e    = (int)((blockIdx.x * 256u + threadIdx.x) >> 5);
    int lane = threadIdx.x & 31;
    if (e >= nedges) return;

    int   r = rows[e];
    int   c = cols[e];
    float v = vals[e];

    const float4* hp = (const float4*)(H + (size_t)c * NFEAT);
    float4 m = hp[lane];

    float* out = AGG + (size_t)r * NFEAT + (size_t)lane * 4;
    unsafeAtomicAdd(out + 0, m.x * v);
    unsafeAtomicAdd(out + 1, m.y * v);
    unsafeAtomicAdd(out + 2, m.z * v);
    unsafeAtomicAdd(out + 3, m.w * v);
}

// ---------------------------------------------------------------------------
__global__ __launch_bounds__(256) void gcn_zero(float4* __restrict__ p, int n4)
{
    int i = blockIdx.x * 256 + threadIdx.x;
    if (i < n4) {
        float4 z; z.x = 0.f; z.y = 0.f; z.z = 0.f; z.w = 0.f;
        p[i] = z;
    }
}

__global__ __launch_bounds__(256) void gcn_elu(
    const float* __restrict__ in, float* __restrict__ out, int n)
{
    int i = blockIdx.x * 256 + threadIdx.x;
    if (i < n) {
        float v = in[i];
        out[i] = (v > 0.0f) ? v : (__expf(v) - 1.0f);
    }
}

// ---------------------------------------------------------------------------
extern "C" void kernel_launch(void* const* d_in, const int* in_sizes, int n_in,
                              void* d_out, int out_size, void* d_ws, size_t ws_size,
                              hipStream_t stream)
{
    const float* x    = (const float*)d_in[0];  // [N, 128]
    const float* W    = (const float*)d_in[1];  // [L, 128, 128]
    const float* b    = (const float*)d_in[2];  // [L, 128]
    const int*   rows = (const int*)  d_in[3];  // [L, E]
    const int*   cols = (const int*)  d_in[4];  // [L, E]
    const float* vals = (const float*)d_in[5];  // [L, E]

    const int nnodes  = in_sizes[0] / NFEAT;
    const int nlayers = in_sizes[1] / (NFEAT * NFEAT);
    const int nedges  = in_sizes[3] / nlayers;

    const size_t nodeElems = (size_t)nnodes * NFEAT;
    float* hbuf = (float*)d_ws;          // [N,128] linear output
    float* agg  = hbuf + nodeElems;      // [N,128] aggregation / activation

    const int n          = (int)nodeElems;
    const int gemmBlocks = (nnodes + 15) / 16;
    const int zeroBlocks = (n / 4 + 255) / 256;
    const int eluBlocks  = (n + 255) / 256;
    const int scatBlocks = (nedges + 7) / 8;   // 8 edges (waves) per block

    for (int l = 0; l < nlayers; ++l) {
        const float* cur = (l == 0) ? x : agg;   // prev activation
        gcn_gemm_bias<<<gemmBlocks, 256, 0, stream>>>(
            cur, W + (size_t)l * NFEAT * NFEAT, b + (size_t)l * NFEAT,
            hbuf, nnodes);

        gcn_zero<<<zeroBlocks, 256, 0, stream>>>((float4*)agg, n / 4);

        gcn_scatter<<<scatBlocks, 256, 0, stream>>>(
            hbuf, rows + (size_t)l * nedges, cols + (size_t)l * nedges,
            vals + (size_t)l * nedges, agg, nedges);

        float* eluOut = (l == nlayers - 1) ? (float*)d_out : agg;  // in-place ok
        gcn_elu<<<eluBlocks, 256, 0, stream>>>(agg, eluOut, n);
    }
}